// SpGraphAttentionLayer_3195455668441
// MI455X (gfx1250) — compile-verified
//
#include <hip/hip_runtime.h>
#include <math.h>

#define ALPHA 0.2f
#define FIN   512
#define FOUT  256

typedef __attribute__((ext_vector_type(2))) float v2f;
typedef __attribute__((ext_vector_type(8))) float v8f;

// ---------------------------------------------------------------------------
// Kernel 1: h = node @ W   (fp32 WMMA 16x16x4, double-buffered LDS,
//                           fully async-to-LDS staging for A and B tiles)
// Block: 256 threads = 8 waves. Block tile: 64(M) x 256(N).
// Wave (wm,wn): wm in 0..3 -> 16-row slab, wn in 0..1 -> 128-col slab.
// Each wave: 8 accumulators of 16x16 f32 (v8f each).
// ---------------------------------------------------------------------------
#define KC 32
#define MB 64
// A row stride 36 floats: 36*row mod 64 hits 16 distinct even banks (period
// 16, spacing 4) -> conflict-free b64 K-pair fragment loads, and byte stride
// 144 keeps every float4 staging destination 16B-aligned for async B128.
#define A_STRIDE 36

// Per-lane async copy of 16 bytes global -> LDS (CDNA5 ASYNCcnt path).
__device__ __forceinline__ void async_copy_b128(void* lds_dst, const void* gsrc)
{
    unsigned int lds = (unsigned int)(size_t)lds_dst;   // LDS byte offset (low 32 bits)
    asm volatile("global_load_async_to_lds_b128 %0, %1, off"
                 :: "v"(lds), "v"(gsrc)
                 : "memory");
}

__device__ __forceinline__ void wait_asynccnt0()
{
    asm volatile("s_wait_asynccnt 0x0" ::: "memory");
}

__global__ __launch_bounds__(256) void gat_gemm_h(
    const float* __restrict__ node, const float* __restrict__ W,
    float* __restrict__ h, int N)
{
    __shared__ float Asl[2][MB * A_STRIDE];   // 2 x (64 x 32, stride 36) ~18.4 KB
    __shared__ float Bsl[2][KC * FOUT];       // 2 x (32 x 256)           ~65.5 KB

    const int tid    = threadIdx.x;
    const int lane   = tid & 31;
    const int wave   = tid >> 5;
    const int wm     = wave & 3;
    const int wn     = wave >> 2;
    const int m0     = blockIdx.x * MB;
    const int lane15 = lane & 15;
    const int koff   = (lane >> 4) << 1;        // 0 (lanes 0-15) or 2 (16-31)
    const int arow   = wm * 16 + lane15;        // A-frag row within block tile

    v8f acc[8] = {};

    // ---- staging helpers: all-async (ASYNCcnt-tracked) -------------------
    auto stage_A = [&](int k0, int buf) {       // 64x32 chunk of node
        #pragma unroll
        for (int i = 0; i < 2; ++i) {
            int f4 = tid + i * 256;              // 0..511
            int r  = f4 >> 3;                    // 8 float4 per row
            int c4 = f4 & 7;
            int gr = m0 + r; if (gr >= N) gr = N - 1;   // clamp tail reads
            async_copy_b128(&Asl[buf][r * A_STRIDE + c4 * 4],
                            node + (size_t)gr * FIN + k0 + c4 * 4);
        }
    };
    auto stage_B = [&](int k0, int buf) {       // 32x256 chunk of W
        #pragma unroll
        for (int i = 0; i < 8; ++i) {
            int f4 = tid + i * 256;              // 0..2047
            int r  = f4 >> 6;                    // 64 float4 per row
            int c4 = f4 & 63;
            async_copy_b128(&Bsl[buf][r * FOUT + c4 * 4],
                            W + (size_t)(k0 + r) * FOUT + c4 * 4);
        }
    };

    // ---- prologue: fill buffer 0 ----------------------------------------
    stage_B(0, 0);
    stage_A(0, 0);
    wait_asynccnt0();
    __syncthreads();

    const int NCHUNK = FIN / KC;                 // 16
    for (int c = 0; c < NCHUNK; ++c) {
        const int cur = c & 1;
        const int nxt = cur ^ 1;
        if (c + 1 < NCHUNK) {                    // prefetch next chunk (async)
            stage_B((c + 1) * KC, nxt);
            stage_A((c + 1) * KC, nxt);
        }

        // ---- 8 K-steps of 4, 8 WMMAs each, on current buffer ----
        const float* Ab = Asl[cur];
        const float* Bb = Bsl[cur];
        #pragma unroll
        for (int ks = 0; ks < KC; ks += 4) {
            const int ka = ks + koff;
            // aligned, bank-conflict-free b64 load of the A K-pair
            const v2f a = *(const v2f*)(&Ab[arow * A_STRIDE + ka]);
            #pragma unroll
            for (int nt = 0; nt < 8; ++nt) {
                const int col = wn * 128 + nt * 16 + lane15;
                v2f b;
                b.x = Bb[ka * FOUT + col];
                b.y = Bb[(ka + 1) * FOUT + col];
                acc[nt] = __builtin_amdgcn_wmma_f32_16x16x4_f32(
                    false, a, false, b, (short)0, acc[nt], false, false);
            }
        }

        wait_asynccnt0();                        // next-buffer async copies done
        __syncthreads();                         // make them visible block-wide
    }

    // ---- store D: VGPR v -> row v (lanes 0-15) / row v+8 (lanes 16-31) ----
    const int rbase = m0 + wm * 16 + ((lane >> 4) << 3);
    for (int nt = 0; nt < 8; ++nt) {
        const int col = wn * 128 + nt * 16 + lane15;
        #pragma unroll
        for (int v = 0; v < 8; ++v) {
            const int row = rbase + v;
            if (row < N) h[(size_t)row * FOUT + col] = acc[nt][v];
        }
    }
}

// ---------------------------------------------------------------------------
// Kernel 2: s_src[n] = h[n]·a[0:256], s_dst[n] = h[n]·a[256:512]
// One wave32 per node, shuffle reduction.
// ---------------------------------------------------------------------------
__global__ __launch_bounds__(256) void gat_scores(
    const float* __restrict__ h, const float* __restrict__ a,
    float* __restrict__ s_src, float* __restrict__ s_dst, int N)
{
    const int lane = threadIdx.x & 31;
    const int n    = blockIdx.x * 8 + (threadIdx.x >> 5);
    if (n >= N) return;
    float ss = 0.f, sd = 0.f;
    #pragma unroll
    for (int i = 0; i < FOUT; i += 32) {
        const float hv = h[(size_t)n * FOUT + i + lane];
        ss += hv * a[i + lane];
        sd += hv * a[FOUT + i + lane];
    }
    #pragma unroll
    for (int off = 16; off > 0; off >>= 1) {
        ss += __shfl_xor(ss, off, 32);
        sd += __shfl_xor(sd, off, 32);
    }
    if (lane == 0) { s_src[n] = ss; s_dst[n] = sd; }
}

// ---------------------------------------------------------------------------
// Kernel 3: per-edge attention coefficient + rowsum accumulation.
// edge_e written directly into output region 2 of d_out.
// ---------------------------------------------------------------------------
__global__ void gat_edge(
    const long long* __restrict__ edge, const float* __restrict__ ef,
    const float* __restrict__ s_src, const float* __restrict__ s_dst,
    const float* __restrict__ a, float* __restrict__ edge_e,
    float* __restrict__ rowsum, int E)
{
    const int e = blockIdx.x * blockDim.x + threadIdx.x;
    if (e >= E) return;
    const long long src = edge[2 * e];
    const long long dst = edge[2 * e + 1];
    const float logit = s_src[src] + s_dst[dst] + ef[e] * a[2 * FOUT];
    const float lr = logit > 0.f ? logit : ALPHA * logit;
    const float ee = expf(-lr);
    edge_e[e] = ee;
    atomicAdd(&rowsum[src], ee);
}

// ---------------------------------------------------------------------------
// Kernel 4: h_prime[src] += edge_e * h[dst]   (L2-resident scatter)
// One wave32 per edge; lane covers 8 columns; f32 global atomics.
// ---------------------------------------------------------------------------
__global__ __launch_bounds__(256) void gat_scatter(
    const long long* __restrict__ edge, const float* __restrict__ edge_e,
    const float* __restrict__ h, float* __restrict__ hprime, int E)
{
    const int lane = threadIdx.x & 31;
    const int e    = blockIdx.x * 8 + (threadIdx.x >> 5);
    if (e >= E) return;
    const long long src = edge[2 * e];
    const long long dst = edge[2 * e + 1];
    const float ee = edge_e[e];
    const float* hd = h + (size_t)dst * FOUT;
    float* hp = hprime + (size_t)src * FOUT;
    #pragma unroll
    for (int i = 0; i < FOUT; i += 32)
        atomicAdd(&hp[i + lane], ee * hd[i + lane]);
}

// ---------------------------------------------------------------------------
// Kernel 5: h_prime/rowsum (NaN->0), residual, ELU -> d_out region 1.
// ---------------------------------------------------------------------------
__global__ void gat_final(
    const float* __restrict__ h, const float* __restrict__ hprime,
    const float* __restrict__ rowsum, float* __restrict__ out, long long total)
{
    const long long idx = (long long)blockIdx.x * blockDim.x + threadIdx.x;
    if (idx >= total) return;
    const int n = (int)(idx >> 8);           // FOUT == 256
    const float rs = rowsum[n];
    float v = rs > 0.f ? hprime[idx] / rs : 0.f;   // 0/0 -> NaN -> 0 semantics
    v += h[idx];
    out[idx] = v > 0.f ? v : expf(v) - 1.f;  // ELU(alpha=1)
}

__global__ void gat_zero(float* __restrict__ p, long long n)
{
    const long long i = (long long)blockIdx.x * blockDim.x + threadIdx.x;
    if (i < n) p[i] = 0.f;
}

// ---------------------------------------------------------------------------
extern "C" void kernel_launch(void* const* d_in, const int* in_sizes, int n_in,
                              void* d_out, int out_size, void* d_ws, size_t ws_size,
                              hipStream_t stream)
{
    const float*     node = (const float*)d_in[0];
    const long long* edge = (const long long*)d_in[1];
    const float*     ef   = (const float*)d_in[2];
    const float*     W    = (const float*)d_in[3];
    const float*     a    = (const float*)d_in[4];

    const int N  = in_sizes[0] / FIN;
    const int E  = in_sizes[1] / 2;
    const long long NF = (long long)N * FOUT;

    // workspace layout (floats): h | hprime | rowsum | s_src | s_dst
    float* h      = (float*)d_ws;
    float* hprime = h + NF;
    float* rowsum = hprime + NF;
    float* s_src  = rowsum + N;
    float* s_dst  = s_src + N;

    float* out    = (float*)d_out;       // region 1: N*FOUT
    float* edge_e = out + NF;            // region 2: E

    // zero accumulators (hprime and rowsum are contiguous)
    {
        const long long n = NF + N;
        gat_zero<<<(unsigned)((n + 255) / 256), 256, 0, stream>>>(hprime, n);
    }
    gat_gemm_h<<<(unsigned)((N + MB - 1) / MB), 256, 0, stream>>>(node, W, h, N);
    gat_scores<<<(unsigned)((N + 7) / 8), 256, 0, stream>>>(h, a, s_src, s_dst, N);
    gat_edge<<<(unsigned)((E + 255) / 256), 256, 0, stream>>>(
        edge, ef, s_src, s_dst, a, edge_e, rowsum, E);
    gat_scatter<<<(unsigned)((E + 7) / 8), 256, 0, stream>>>(
        edge, edge_e, h, hprime, E);
    gat_final<<<(unsigned)((NF + 255) / 256), 256, 0, stream>>>(
        h, hprime, rowsum, out, NF);
}